// prox_softmax_70300024701180
// MI455X (gfx1250) — compile-verified
//
#include <hip/hip_runtime.h>

#define TOL    1e-4f
#define LOG2E  1.4426950408889634f

constexpr int ROW    = 4096;
constexpr int TPB    = 256;
constexpr int VEC    = 4;
constexpr int EPT    = ROW / TPB;    // 16 elements per thread
constexpr int CHUNK  = TPB * VEC;    // 1024 floats per load sweep
constexpr int NCHUNK = ROW / CHUNK;  // 4
constexpr int NWAVE  = TPB / 32;     // 8 wave32's per block

// v_exp_f32 path: exp(x) = exp2(x * log2(e))
__device__ __forceinline__ float fast_exp(float x) {
    return __builtin_amdgcn_exp2f(x * LOG2E);
}
// v_rcp_f32 (1 ulp) instead of IEEE division sequence
__device__ __forceinline__ float fast_rcp(float x) {
    return __builtin_amdgcn_rcpf(x);
}

// Generic LDS pointer: aperture in high 32 bits, LDS byte offset in low 32.
__device__ __forceinline__ unsigned lds_lo32(const void* p) {
    return (unsigned)(unsigned long long)p;
}

// wave32 xor-butterfly step via ds_swizzle_b32 (group-of-32 mode:
// and_mask=0x1f, or_mask=0, xor_mask=offset[14:10])
template <int XORM>
__device__ __forceinline__ float swz_xor(float v) {
    constexpr int IMM = (XORM << 10) | 0x1F;
    return __int_as_float(__builtin_amdgcn_ds_swizzle(__float_as_int(v), IMM));
}
__device__ __forceinline__ float wave_sum(float v) {
    v += swz_xor<1>(v);  v += swz_xor<2>(v);  v += swz_xor<4>(v);
    v += swz_xor<8>(v);  v += swz_xor<16>(v);
    return v;
}
__device__ __forceinline__ float wave_max(float v) {
    v = fmaxf(v, swz_xor<1>(v));  v = fmaxf(v, swz_xor<2>(v));
    v = fmaxf(v, swz_xor<4>(v));  v = fmaxf(v, swz_xor<8>(v));
    v = fmaxf(v, swz_xor<16>(v));
    return v;
}

// One full-row evaluation at the current nu:
//   Halley-solve l_i (NH iterations, COLD => analytic start, else warm start
//   from previous l), produce p_i = exp(l_i), and the two row partials
//   sp = sum p, sd = sum p/(u - 0.5 l).  alpha = 0.5 folded in:
//   F = 2*resid = (2u-1) - x - e^x, dF = -1-e, ddF = -e.
template <int NH, bool COLD>
__device__ __forceinline__ void eval_row(const float (&y)[EPT], float (&l)[EPT],
                                         float (&p)[EPT], float nu,
                                         float& sp, float& sd) {
    sp = 0.f; sd = 0.f;
#pragma unroll
    for (int k = 0; k < EPT; ++k) {
        const float u = y[k] + nu;
        const float c = 2.f * u - 1.f;
        float xv = COLD ? fminf(c, u - 1.f) : l[k];
#pragma unroll
        for (int h = 0; h < NH; ++h) {
            const float e   = fast_exp(xv);
            const float F   = c - xv - e;
            const float dF  = -1.f - e;
            const float num = 2.f * F * dF;
            const float den = fmaf(2.f * dF, dF, F * e);   // 2dF^2 - F*ddF
            xv -= num * fast_rcp(den);
        }
        l[k] = xv;
        const float pk = fast_exp(xv);
        p[k] = pk;
        sp += pk;
        sd += pk * fast_rcp(u - 0.5f * xv);
    }
}

__global__ __launch_bounds__(TPB)
void prox_softmax_kernel(const float* __restrict__ x,
                         float* __restrict__ out, int nrows) {
    __shared__ float row_s[ROW];     // 16 KB row staging (of 320 KB/WGP)
    __shared__ float red_a[NWAVE];
    __shared__ float red_b[NWAVE];

    const int r = blockIdx.x;
    if (r >= nrows) return;
    const int tid  = threadIdx.x;
    const int lane = tid & 31;
    const int wave = tid >> 5;
    const float* __restrict__ grow = x + (size_t)r * ROW;

    // ---- CDNA5 async global -> LDS load of the whole row (ASYNCcnt) ----
#pragma unroll
    for (int c = 0; c < NCHUNK; ++c) {
        const int idx = c * CHUNK + tid * VEC;
        unsigned lds = lds_lo32(&row_s[idx]);
        const float* g = grow + idx;
        asm volatile("global_load_async_to_lds_b128 %0, %1, off"
                     :: "v"(lds), "v"(g) : "memory");
    }
    asm volatile("s_wait_asynccnt 0x0" ::: "memory");
    __syncthreads();

    // LDS -> registers (ds_load_b128); row lives in VGPRs from here on.
    float y[EPT];
#pragma unroll
    for (int c = 0; c < NCHUNK; ++c) {
        const float4 v =
            *reinterpret_cast<const float4*>(&row_s[c * CHUNK + tid * VEC]);
        y[c * VEC + 0] = v.x; y[c * VEC + 1] = v.y;
        y[c * VEC + 2] = v.z; y[c * VEC + 3] = v.w;
    }

    // nu0 = 1 - max(y) over the row
    float m = y[0];
#pragma unroll
    for (int k = 1; k < EPT; ++k) m = fmaxf(m, y[k]);
    m = wave_max(m);
    if (lane == 0) red_a[wave] = m;
    __syncthreads();
    float mr = red_a[0];
#pragma unroll
    for (int w = 1; w < NWAVE; ++w) mr = fmaxf(mr, red_a[w]);

    float nu = 1.0f - mr;
    float l[EPT], p[EPT];
    float sp, sd;

    // Cold-start evaluation (8 Halley iters, analytic x0)
    eval_row<8, true>(y, l, p, nu, sp, sd);

    // Outer Newton on nu: sum_i exp(l_i) = 1
    for (int it = 0; it < 64; ++it) {
        // two-value block reduction: ds_swizzle butterfly + 8 LDS partials
        sp = wave_sum(sp);
        sd = wave_sum(sd);
        __syncthreads();
        if (lane == 0) { red_a[wave] = sp; red_b[wave] = sd; }
        __syncthreads();
        float fsum = 0.f, dsum = 0.f;
#pragma unroll
        for (int w = 0; w < NWAVE; ++w) { fsum += red_a[w]; dsum += red_b[w]; }

        const float f = fsum - 1.0f;          // uniform across block
        if (fabsf(f) < TOL) break;
        nu -= f * fast_rcp(dsum);

        // Warm-start re-evaluation (3 Halley iters from previous l)
        eval_row<3, false>(y, l, p, nu, sp, sd);
    }

    // p already in registers from the converged evaluation
    float* __restrict__ orow = out + (size_t)r * ROW;
#pragma unroll
    for (int c = 0; c < NCHUNK; ++c) {
        const int idx = c * CHUNK + tid * VEC;
        float4 v;
        v.x = p[c * VEC + 0]; v.y = p[c * VEC + 1];
        v.z = p[c * VEC + 2]; v.w = p[c * VEC + 3];
        *reinterpret_cast<float4*>(orow + idx) = v;
    }
}

extern "C" void kernel_launch(void* const* d_in, const int* in_sizes, int n_in,
                              void* d_out, int out_size, void* d_ws, size_t ws_size,
                              hipStream_t stream) {
    const float* x = (const float*)d_in[0];
    float* out = (float*)d_out;
    const int nrows = in_sizes[0] / ROW;   // 8192 rows of 4096
    prox_softmax_kernel<<<dim3(nrows), dim3(TPB), 0, stream>>>(x, out, nrows);
}